// Graphsage_51084341018874
// MI455X (gfx1250) — compile-verified
//
#include <hip/hip_runtime.h>
#include <math.h>

#define C 64
#define CP 65                    // padded LDS stride (bank-conflict free)
#define ROWS 64                  // rows per block (4 sub-tiles of 16)
#define EPS_NORM 1e-12f
#define EPS_BN   1e-5f

typedef float v2f __attribute__((ext_vector_type(2)));
typedef float v8f __attribute__((ext_vector_type(8)));

// ---------------- utility kernels ----------------

__global__ void zero_f4_kernel(float4* __restrict__ p, long long n4) {
    long long i = (long long)blockIdx.x * blockDim.x + threadIdx.x;
    if (i < n4) p[i] = make_float4(0.f, 0.f, 0.f, 0.f);
}

__global__ void degree_kernel(const int* __restrict__ dst, float* __restrict__ deg, int nE) {
    int e = blockIdx.x * blockDim.x + threadIdx.x;
    if (e < nE) atomicAdd(&deg[dst[e]], 1.0f);
}

__global__ void invdeg_kernel(const float* __restrict__ deg, float* __restrict__ inv, int n) {
    int i = blockIdx.x * blockDim.x + threadIdx.x;
    if (i < n) inv[i] = 1.0f / fmaxf(deg[i], 1.0f);
}

// one thread per (edge, 4 channels): float4 (b128) coalesced gather,
// 4x no-return global_atomic_add_f32 scatter; h/agg are L2-resident (25.6MB each).
__global__ void scatter_kernel(const float* __restrict__ h,
                               const int* __restrict__ src,
                               const int* __restrict__ dst,
                               float* __restrict__ agg, int nE) {
    long long t = (long long)blockIdx.x * blockDim.x + threadIdx.x;
    int e  = (int)(t >> 4);
    int c4 = (int)(t & 15) * 4;
    if (e < nE) {
        int s = src[e];
        int d = dst[e];
        float4 v = *(const float4*)(h + (long long)s * C + c4);
        float* a = agg + (long long)d * C + c4;
        atomicAdd(a + 0, v.x);
        atomicAdd(a + 1, v.y);
        atomicAdd(a + 2, v.z);
        atomicAdd(a + 3, v.w);
    }
}

// ---------------- fused SAGE layer: WMMA GEMM + bias + L2norm + BN + ReLU ----

__global__ __launch_bounds__(128)
void sage_gemm_kernel(const float* __restrict__ h,      // [n, 64] input features
                      const float* __restrict__ agg,    // [n, 64] summed neighbors
                      const float* __restrict__ invdeg, // [n]
                      const float* __restrict__ w1,     // [64, 64] row-major
                      const float* __restrict__ w2,     // [64, 64]
                      const float* __restrict__ b2,     // [64]
                      const float* __restrict__ bng, const float* __restrict__ bnb,
                      const float* __restrict__ bnm, const float* __restrict__ bnv,
                      float* __restrict__ out,          // [n, 64]
                      int n, int doBnRelu)
{
    __shared__ float sW1[C * CP];        // sW1[j*CP+k] = w1[j][k]  (B fragment: B[k][j])
    __shared__ float sW2[C * CP];
    __shared__ float sA [ROWS * CP];     // h tile (64 rows)
    __shared__ float sN [ROWS * CP];     // mean-neighbor tile (agg * invdeg)
    __shared__ float sO [ROWS * CP];     // output tile
    __shared__ float sScale[ROWS];

    const int tid = threadIdx.x;
    const int m0  = blockIdx.x * ROWS;

    // stage weights (16KB each) into LDS, coalesced — amortized over 64 rows
    for (int i = tid; i < C * C; i += 128) {
        int j = i >> 6, k = i & 63;
        sW1[j * CP + k] = w1[i];
        sW2[j * CP + k] = w2[i];
    }
    // stage 64-row feature tiles; compute mean neighbors on the fly
    for (int i = tid; i < ROWS * C; i += 128) {
        int r = i >> 6, c = i & 63;
        int row = m0 + r;
        if (row >= n) row = n - 1;       // clamp tail block
        float id = invdeg[row];
        sA[r * CP + c] = h[(long long)row * C + c];
        sN[r * CP + c] = agg[(long long)row * C + c] * id;
    }
    __syncthreads();

    const int lane = tid & 31;
    const int wv   = tid >> 5;           // wave id -> N-tile [16*wv, 16*wv+16)
    const int mn   = lane & 15;          // A: row m; B: col n (within tile)
    const int half = lane >> 4;          // K sub-phase per documented 16x4 f32 layout
    const int jcol = wv * 16 + mn;       // output column handled by this lane

    // preload this wave's B fragments for both weights into registers (64 VGPRs)
    v2f fb1[16], fb2[16];
#pragma unroll
    for (int kk = 0; kk < 16; ++kk) {
        int ka = kk * 4 + half * 2;
        fb1[kk].x = sW1[jcol * CP + ka];
        fb1[kk].y = sW1[jcol * CP + ka + 1];
        fb2[kk].x = sW2[jcol * CP + ka];
        fb2[kk].y = sW2[jcol * CP + ka + 1];
    }

    const float bias = b2[jcol];

    // 4 sub-tiles of 16 rows; 32 chained v_wmma_f32_16x16x4_f32 each
    for (int t = 0; t < 4; ++t) {
        const int r0 = t * 16;
        v8f acc = {};
#pragma unroll
        for (int kk = 0; kk < 16; ++kk) {
            int ka = kk * 4 + half * 2;
            v2f a;
            a.x = sA[(r0 + mn) * CP + ka];
            a.y = sA[(r0 + mn) * CP + ka + 1];
            acc = __builtin_amdgcn_wmma_f32_16x16x4_f32(false, a, false, fb1[kk],
                                                        (short)0, acc, false, false);
        }
#pragma unroll
        for (int kk = 0; kk < 16; ++kk) {
            int ka = kk * 4 + half * 2;
            v2f a;
            a.x = sN[(r0 + mn) * CP + ka];
            a.y = sN[(r0 + mn) * CP + ka + 1];
            acc = __builtin_amdgcn_wmma_f32_16x16x4_f32(false, a, false, fb2[kk],
                                                        (short)0, acc, false, false);
        }
        // bias + spill 16x16 tile (C/D layout: VGPR v -> row half*8+v, col lane&15)
#pragma unroll
        for (int v = 0; v < 8; ++v) {
            int r = r0 + half * 8 + v;
            sO[r * CP + jcol] = acc[v] + bias;
        }
    }
    __syncthreads();

    // per-row L2 norm across all 64 columns (one thread per row)
    if (tid < ROWS) {
        float s = 0.f;
        for (int c = 0; c < C; ++c) {
            float x = sO[tid * CP + c];
            s += x * x;
        }
        sScale[tid] = 1.0f / fmaxf(sqrtf(s), EPS_NORM);
    }
    __syncthreads();

    // normalize + optional BN(eval)+ReLU, write next-layer features
    for (int i = tid; i < ROWS * C; i += 128) {
        int r = i >> 6, c = i & 63;
        int row = m0 + r;
        if (row < n) {
            float x = sO[r * CP + c] * sScale[r];
            if (doBnRelu) {
                x = bng[c] * (x - bnm[c]) * rsqrtf(bnv[c] + EPS_BN) + bnb[c];
                x = fmaxf(x, 0.f);
            }
            out[(long long)row * C + c] = x;
        }
    }
}

// ---------------- launcher ----------------

extern "C" void kernel_launch(void* const* d_in, const int* in_sizes, int n_in,
                              void* d_out, int out_size, void* d_ws, size_t ws_size,
                              hipStream_t stream) {
    const float* x    = (const float*)d_in[0];
    const int*   esrc = (const int*)d_in[1];
    const int*   edst = (const int*)d_in[2];
    // setup_inputs() dict order: w1_i, w2_i, b2_i for i=0..2, then bn_{g,b,m,v}_i for i=0..1
    const float* w1[3]  = { (const float*)d_in[3], (const float*)d_in[6], (const float*)d_in[9]  };
    const float* w2[3]  = { (const float*)d_in[4], (const float*)d_in[7], (const float*)d_in[10] };
    const float* b2[3]  = { (const float*)d_in[5], (const float*)d_in[8], (const float*)d_in[11] };
    const float* bng[2] = { (const float*)d_in[12], (const float*)d_in[16] };
    const float* bnb[2] = { (const float*)d_in[13], (const float*)d_in[17] };
    const float* bnm[2] = { (const float*)d_in[14], (const float*)d_in[18] };
    const float* bnv[2] = { (const float*)d_in[15], (const float*)d_in[19] };

    const int n  = in_sizes[0] / C;   // 100000
    const int nE = in_sizes[1];       // 1600000

    // workspace carve (floats): deg[n] | inv[n] | agg[n*C] | h1[n*C] | h2[n*C]
    float* ws  = (float*)d_ws;
    float* deg = ws;
    float* inv = deg + n;
    float* agg = inv + n;
    float* h1  = agg + (size_t)n * C;
    float* h2  = h1 + (size_t)n * C;

    const int ZT = 256;

    // degrees (identical for all layers): zero deg+inv, count, invert
    {
        long long n4 = ((long long)2 * n + 3) / 4;
        zero_f4_kernel<<<(unsigned)((n4 + ZT - 1) / ZT), ZT, 0, stream>>>((float4*)deg, n4);
        degree_kernel<<<(nE + ZT - 1) / ZT, ZT, 0, stream>>>(edst, deg, nE);
        invdeg_kernel<<<(n + ZT - 1) / ZT, ZT, 0, stream>>>(deg, inv, n);
    }

    const float* hin[3]  = { x, h1, h2 };
    float*       hout[3] = { h1, h2, (float*)d_out };

    long long aggN4 = ((long long)n * C) / 4;
    long long scatterThreads = (long long)nE * (C / 4);
    unsigned scatterBlocks = (unsigned)((scatterThreads + ZT - 1) / ZT);
    unsigned gemmBlocks = (unsigned)((n + ROWS - 1) / ROWS);

    for (int l = 0; l < 3; ++l) {
        zero_f4_kernel<<<(unsigned)((aggN4 + ZT - 1) / ZT), ZT, 0, stream>>>((float4*)agg, aggN4);
        scatter_kernel<<<scatterBlocks, ZT, 0, stream>>>(hin[l], esrc, edst, agg, nE);
        int doBn = (l < 2) ? 1 : 0;
        int bi = (l < 2) ? l : 0;     // dummy (unused) bn pointers for last layer
        sage_gemm_kernel<<<gemmBlocks, 128, 0, stream>>>(
            hin[l], agg, inv, w1[l], w2[l], b2[l],
            bng[bi], bnb[bi], bnm[bi], bnv[bi],
            hout[l], n, doBn);
    }
}